// Seq2SetLoss_56049323212920
// MI455X (gfx1250) — compile-verified
//
#include <hip/hip_runtime.h>
#include <math.h>

// Problem constants (match reference)
#define BB  128
#define LL  48
#define VV  32000
#define NBR 16

typedef float v2f __attribute__((ext_vector_type(2)));
typedef float v4f __attribute__((ext_vector_type(4)));
typedef float v8f __attribute__((ext_vector_type(8)));

// ---------------------------------------------------------------------------
// Wave32 sum-reduction via two chained V_WMMA_F32_16X16X4_F32 with B = ones.
// Stage 1: A[m][0]=p_m (lanes 0-15, VGPR0), A[m][2]=p_{m+16} (lanes 16-31,
//          VGPR0), K1/K3 = 0  ->  D[m][n] = p_m + p_{m+16}  (replicated in n).
// Stage 2: each lane folds its 8 D VGPRs into 2 A values; the second WMMA
//          sums the remaining 16 partials -> every element of D2 equals the
//          total of all 32 lane inputs (broadcast, deterministic).
// Correct regardless of B's exact VGPR layout because B is constant 1.
// ---------------------------------------------------------------------------
__device__ __forceinline__ float wave_sum_wmma(float v) {
    v2f a;    a[0] = v;    a[1] = 0.0f;
    v2f ones; ones[0] = 1.0f; ones[1] = 1.0f;
    v8f c = {0.f, 0.f, 0.f, 0.f, 0.f, 0.f, 0.f, 0.f};
    v8f d = __builtin_amdgcn_wmma_f32_16x16x4_f32(
        false, a, false, ones, (short)0, c, false, false);
    v2f a2;
    a2[0] = d[0] + d[1] + d[2] + d[3];
    a2[1] = d[4] + d[5] + d[6] + d[7];
    v8f d2 = __builtin_amdgcn_wmma_f32_16x16x4_f32(
        false, a2, false, ones, (short)0, c, false, false);
    return d2[0];
}

// ---------------------------------------------------------------------------
// Kernel 1: one workgroup per (b,l) row of 32000 logits.
// Streams the row once (non-temporal b128 loads), online softmax:
//   nll = max + log(sum exp(x - max)) - x[target]
// Memory-bound: 786 MB total -> ~34us floor at 23.3 TB/s.
// ---------------------------------------------------------------------------
__global__ __launch_bounds__(256) void row_nll_kernel(
    const float* __restrict__ pred, const int* __restrict__ tgt,
    float* __restrict__ nll_out, float* __restrict__ val_out) {
    const int r   = blockIdx.x;          // row in [0, B*L)
    const int tid = threadIdx.x;
    const int t   = tgt[r];
    const int tchunk = (t >= 0) ? (t >> 2) : -1;

    const v4f* p4 = reinterpret_cast<const v4f*>(pred + (size_t)r * VV);

    float m = -INFINITY, s = 0.0f, xt = 0.0f;
    for (int i = tid; i < VV / 4; i += 256) {
        v4f v = __builtin_nontemporal_load(&p4[i]);   // stream-once: NT hint
        float m4 = fmaxf(fmaxf(v[0], v[1]), fmaxf(v[2], v[3]));
        float mn = fmaxf(m, m4);
        s = s * __expf(m - mn)
            + __expf(v[0] - mn) + __expf(v[1] - mn)
            + __expf(v[2] - mn) + __expf(v[3] - mn);
        m = mn;
        if (i == tchunk) {
            int j = t & 3;
            xt = (j == 0) ? v[0] : (j == 1) ? v[1] : (j == 2) ? v[2] : v[3];
        }
    }

    // wave-level max (5-step xor shuffle, wave32)
    float wm = m;
    #pragma unroll
    for (int off = 16; off > 0; off >>= 1)
        wm = fmaxf(wm, __shfl_xor(wm, off, 32));

    __shared__ float smax[8], ssum[8], sxt[8];
    const int wid = tid >> 5, lane = tid & 31;
    if (lane == 0) smax[wid] = wm;
    __syncthreads();

    float M = smax[0];
    #pragma unroll
    for (int k = 1; k < 8; ++k) M = fmaxf(M, smax[k]);

    // rescale local partial sum to the global max, then WMMA wave reduction
    float sp = s * __expf(m - M);
    float S  = wave_sum_wmma(sp);   // sum of exp over this wave's elements
    float X  = wave_sum_wmma(xt);   // exactly one lane holds x[t]; rest are 0

    if (lane == 0) { ssum[wid] = S; sxt[wid] = X; }
    __syncthreads();

    if (tid == 0) {
        float St = 0.f, Xt = 0.f;
        #pragma unroll
        for (int k = 0; k < 8; ++k) { St += ssum[k]; Xt += sxt[k]; }
        if (t < 0) {                        // ignore_index == -100
            nll_out[r] = 0.0f;
            val_out[r] = 0.0f;
        } else {
            nll_out[r] = M + logf(St) - Xt; // -log softmax at target
            val_out[r] = 1.0f;
        }
    }
}

// ---------------------------------------------------------------------------
// Kernel 2: per-sample CE + seq_probs, then segment logsumexp over branches.
// d_out = [loss, seq_probs[0..127]]  (129 f32)
// ---------------------------------------------------------------------------
__global__ __launch_bounds__(128) void finalize_kernel(
    const float* __restrict__ nll, const float* __restrict__ validf,
    const int* __restrict__ seg_ids, float* __restrict__ out) {
    __shared__ float negce[BB];
    __shared__ int   seg[BB];
    __shared__ float bl[NBR];

    const int b = threadIdx.x;            // one thread per sample
    float sum = 0.f, cnt = 0.f;
    for (int l = 0; l < LL; ++l) {
        sum += nll[b * LL + l];
        cnt += validf[b * LL + l];
    }
    float ce = sum / cnt;                 // mean NLL over valid tokens
    out[1 + b] = __expf(-sum);            // seq_prob = exp(-count*ce)
    negce[b] = -ce;
    seg[b]   = seg_ids[b];
    __syncthreads();

    if (b < NBR) {                        // generic segment logsumexp
        float mm = -INFINITY;
        for (int i = 0; i < BB; ++i)
            if (seg[i] == b) mm = fmaxf(mm, negce[i]);
        float ss = 0.f, c = 0.f;
        for (int i = 0; i < BB; ++i)
            if (seg[i] == b) { ss += __expf(negce[i] - mm); c += 1.f; }
        bl[b] = -(logf(ss) + mm - logf(c));
    }
    __syncthreads();

    if (b == 0) {
        float acc = 0.f;
        #pragma unroll
        for (int i = 0; i < NBR; ++i) acc += bl[i];
        out[0] = acc / (float)NBR;
    }
}

// ---------------------------------------------------------------------------
extern "C" void kernel_launch(void* const* d_in, const int* in_sizes, int n_in,
                              void* d_out, int out_size, void* d_ws, size_t ws_size,
                              hipStream_t stream) {
    const int*   tgt  = (const int*)d_in[0];   // [B, L] int32
    const float* pred = (const float*)d_in[1]; // [B, L, V] f32
    const int*   seg  = (const int*)d_in[2];   // [B] int32
    float*       out  = (float*)d_out;         // [1 + B] f32

    float* nll    = (float*)d_ws;              // [B*L]
    float* validf = nll + (BB * LL);           // [B*L]

    row_nll_kernel<<<BB * LL, 256, 0, stream>>>(pred, tgt, nll, validf);
    finalize_kernel<<<1, 128, 0, stream>>>(nll, validf, seg, out);
}